// OnlineTripletMining_45475113730253
// MI455X (gfx1250) — compile-verified
//
#include <hip/hip_runtime.h>
#include <cstdint>

#define DDIM 256
#define TRIPLET_MARGIN 0.01f
#define COS_EPS 1e-8f

typedef float v2f __attribute__((ext_vector_type(2)));
typedef float v8f __attribute__((ext_vector_type(8)));

// ---------------------------------------------------------------------------
// Kernel 1: per-row L2 norms. One wave (32 lanes) per row.
// ---------------------------------------------------------------------------
__global__ void __launch_bounds__(256)
otm_norms(const float* __restrict__ E, float* __restrict__ norms, int N) {
  int row  = blockIdx.x * 8 + (threadIdx.x >> 5);
  int lane = threadIdx.x & 31;
  if (row >= N) return;
  const float* r = E + (size_t)row * DDIM;
  float s = 0.f;
  #pragma unroll
  for (int i = 0; i < DDIM / 32; ++i) {
    float v = r[lane + i * 32];
    s += v * v;
  }
  #pragma unroll
  for (int m = 16; m >= 1; m >>= 1)
    s += __shfl_xor(s, m, 32);
  if (lane == 0) norms[row] = sqrtf(s);
}

// ---------------------------------------------------------------------------
// Async global->LDS copy: 32 x b128 per thread, shared offset immediate
// applies to BOTH the LDS and global address (CDNA5 ISA).
// ---------------------------------------------------------------------------
template <int I, int NLOADS>
struct AsyncLoader {
  static __device__ __forceinline__ void run(unsigned ldsoff, const float* g) {
    asm volatile("global_load_async_to_lds_b128 %0, %1, off offset:%2"
                 :: "v"(ldsoff), "v"(g), "i"(I * 16) : "memory");
    AsyncLoader<I + 1, NLOADS>::run(ldsoff, g);
  }
};
template <int NLOADS>
struct AsyncLoader<NLOADS, NLOADS> {
  static __device__ __forceinline__ void run(unsigned, const float*) {}
};

// ---------------------------------------------------------------------------
// Kernel 2: fused  sim = (E E^T)/max(n_i n_j, eps)  +  hardest pos/neg mining.
// Block = 128 threads = 4 waves; each wave owns 16 rows (A-panel in VGPRs)
// and computes a 16x64 output strip per tile: 4 independent WMMA accumulator
// chains (ILP). Column tiles (64 rows of E) are double-buffered in LDS via
// async-to-LDS copies overlapped with compute (ASYNCcnt + barrier).
// ---------------------------------------------------------------------------
__global__ void __launch_bounds__(128)
otm_mine(const float* __restrict__ E, const int* __restrict__ labels,
         const float* __restrict__ norms, float* __restrict__ rowloss, int N) {
  const int STRIDE = 260;  // 260 % 64 == 4 -> conflict-free b64 reads
  __shared__ __align__(16) float Bt[2][64 * STRIDE];

  const int wave    = threadIdx.x >> 5;
  const int lane    = threadIdx.x & 31;
  const int rowbase = blockIdx.x * 64 + wave * 16;
  const int mrow    = lane & 15;          // A-row within tile
  const int koff    = (lane >> 4) << 1;   // K offset: 0 (lanes 0-15) / 2 (16-31)

  // A fragments: this wave's 16 rows, full K = 256 -> 64 v2f fragments.
  v2f a[64];
  const float* Arow = E + (size_t)(rowbase + mrow) * DDIM;
  #pragma unroll
  for (int kk = 0; kk < 64; ++kk)
    a[kk] = *(const v2f*)(Arow + kk * 4 + koff);

  // C/D layout: VGPR vg holds row (vg + 8*(lane>=16)), column (lane&15).
  const int rsel = rowbase + ((lane >> 4) << 3);
  float ni[8]; int li[8];
  #pragma unroll
  for (int vg = 0; vg < 8; ++vg) {
    ni[vg] = norms[rsel + vg];
    li[vg] = labels[rsel + vg];
  }
  float ap[8], an[8];
  #pragma unroll
  for (int vg = 0; vg < 8; ++vg) { ap[vg] = 0.f; an[vg] = 1.f; }

  // Each thread async-copies half of one staged row: matching global/LDS
  // deltas so the shared instruction offset is valid for both addresses.
  const int r_ld = threadIdx.x >> 1;        // 0..63: staged row
  const int h_ld = threadIdx.x & 1;         // half: 0 / 1 (128 floats each)
  auto issue_tile = [&](int ct, int buf) {
    unsigned ldsoff = (unsigned)(uintptr_t)&Bt[buf][r_ld * STRIDE + h_ld * 128];
    const float* g  = E + (size_t)(ct * 64 + r_ld) * DDIM + h_ld * 128;
    AsyncLoader<0, 32>::run(ldsoff, g);     // 32 x b128 = 512B
  };

  const int T = N / 64;                     // 128 column tiles
  issue_tile(0, 0);                         // prologue

  for (int ct = 0; ct < T; ++ct) {
    const int buf = ct & 1;
    asm volatile("s_wait_asynccnt 0x0" ::: "memory");  // my copies for ct done
    __syncthreads();                                   // everyone's done + prev compute done
    if (ct + 1 < T) issue_tile(ct + 1, buf ^ 1);       // overlap next tile's copies

    v8f acc[4];
    #pragma unroll
    for (int t = 0; t < 4; ++t) acc[t] = v8f{0.f,0.f,0.f,0.f,0.f,0.f,0.f,0.f};

    const float* B0 = &Bt[buf][(lane & 15) * STRIDE + koff];
    #pragma unroll
    for (int kk = 0; kk < 64; ++kk) {
      v2f b0 = *(const v2f*)(B0 + 0 * 16 * STRIDE + kk * 4);
      v2f b1 = *(const v2f*)(B0 + 1 * 16 * STRIDE + kk * 4);
      v2f b2 = *(const v2f*)(B0 + 2 * 16 * STRIDE + kk * 4);
      v2f b3 = *(const v2f*)(B0 + 3 * 16 * STRIDE + kk * 4);
      acc[0] = __builtin_amdgcn_wmma_f32_16x16x4_f32(false, a[kk], false, b0,
                                                     (short)0, acc[0], false, false);
      acc[1] = __builtin_amdgcn_wmma_f32_16x16x4_f32(false, a[kk], false, b1,
                                                     (short)0, acc[1], false, false);
      acc[2] = __builtin_amdgcn_wmma_f32_16x16x4_f32(false, a[kk], false, b2,
                                                     (short)0, acc[2], false, false);
      acc[3] = __builtin_amdgcn_wmma_f32_16x16x4_f32(false, a[kk], false, b3,
                                                     (short)0, acc[3], false, false);
    }

    #pragma unroll
    for (int t = 0; t < 4; ++t) {
      const int   j  = ct * 64 + t * 16 + (lane & 15);
      const float nj = norms[j];
      const int   lj = labels[j];
      #pragma unroll
      for (int vg = 0; vg < 8; ++vg) {
        const int   irow = rsel + vg;
        const float dist = 1.f - acc[t][vg] / fmaxf(ni[vg] * nj, COS_EPS);
        const bool  same = (li[vg] == lj);
        if (same && (irow != j)) ap[vg] = fmaxf(ap[vg], dist);
        if (!same)               an[vg] = fminf(an[vg], dist);
      }
    }
  }

  // Reduce over the 16 columns held by the 16 lanes of each half-wave.
  #pragma unroll
  for (int msk = 8; msk >= 1; msk >>= 1) {
    #pragma unroll
    for (int vg = 0; vg < 8; ++vg) {
      ap[vg] = fmaxf(ap[vg], __shfl_xor(ap[vg], msk, 32));
      an[vg] = fminf(an[vg], __shfl_xor(an[vg], msk, 32));
    }
  }
  if ((lane & 15) == 0) {
    #pragma unroll
    for (int vg = 0; vg < 8; ++vg)
      rowloss[rsel + vg] = fmaxf(0.f, ap[vg] - an[vg] + TRIPLET_MARGIN);
  }
}

// ---------------------------------------------------------------------------
// Kernel 3: mean of per-row hinge losses -> scalar.
// ---------------------------------------------------------------------------
__global__ void __launch_bounds__(256)
otm_reduce(const float* __restrict__ rowloss, float* __restrict__ out, int N) {
  __shared__ float sm[256];
  float s = 0.f;
  for (int i = threadIdx.x; i < N; i += 256) s += rowloss[i];
  sm[threadIdx.x] = s;
  __syncthreads();
  for (int o = 128; o > 0; o >>= 1) {
    if ((int)threadIdx.x < o) sm[threadIdx.x] += sm[threadIdx.x + o];
    __syncthreads();
  }
  if (threadIdx.x == 0) out[0] = sm[0] / (float)N;
}

// ---------------------------------------------------------------------------
extern "C" void kernel_launch(void* const* d_in, const int* in_sizes, int n_in,
                              void* d_out, int out_size, void* d_ws, size_t ws_size,
                              hipStream_t stream) {
  const float* E      = (const float*)d_in[0];
  const int*   labels = (const int*)d_in[1];
  float*       out    = (float*)d_out;
  const int    N      = in_sizes[1];          // 8192

  float* norms   = (float*)d_ws;              // N floats
  float* rowloss = norms + N;                 // N floats

  otm_norms <<<dim3((N + 7) / 8), dim3(256), 0, stream>>>(E, norms, N);
  otm_mine  <<<dim3(N / 64),      dim3(128), 0, stream>>>(E, labels, norms, rowloss, N);
  otm_reduce<<<dim3(1),           dim3(256), 0, stream>>>(rowloss, out, N);
}